// qNMS_3762391351431
// MI455X (gfx1250) — compile-verified
//
#include <hip/hip_runtime.h>
#include <stdint.h>

#define M_ROWS 144
#define N_COLS 576
#define N_ITERS 3
#define BATCH 128
#define DEG 24
#define EDGES (M_ROWS * DEG) /* 3456 */
#define TPB 256
#define CHUNK 96            /* tail columns per staging tile: 96*144 == 4*EDGES */
#define NSEG 8              /* row segments in tail: 144/8 = 18 rows per segment */

// Soft quantizer: levels = linspace(-7, 7, 4); eta = 0.5 -> exp(-2*(x-l)^2)
__device__ __forceinline__ float qsoft(float x) {
  const float L0 = -7.0f, L1 = -7.0f / 3.0f, L2 = 7.0f / 3.0f, L3 = 7.0f;
  float d0 = x - L0, d1 = x - L1, d2 = x - L2, d3 = x - L3;
  float e0 = __expf(-2.0f * d0 * d0);
  float e1 = __expf(-2.0f * d1 * d1);
  float e2 = __expf(-2.0f * d2 * d2);
  float e3 = __expf(-2.0f * d3 * d3);
  float s = e0 + e1 + e2 + e3 + 1e-9f;
  return (L0 * e0 + L1 * e1 + L2 * e2 + L3 * e3) / s;
}

__device__ __forceinline__ float sgnf(float x) {
  return (x > 0.0f) ? 1.0f : ((x < 0.0f) ? -1.0f : 0.0f);
}

// Low 32 bits of a flat LDS pointer == LDS byte offset on gfx1250.
__device__ __forceinline__ uint32_t lds_off(const void* p) {
  return (uint32_t)(uintptr_t)p;
}

// CDNA5 async DMA: global -> LDS, tracked by ASYNCcnt.
__device__ __forceinline__ void async_gather_b32(uint32_t lds, const void* g) {
  asm volatile("global_load_async_to_lds_b32 %0, %1, off"
               :: "v"(lds), "v"((unsigned long long)(uintptr_t)g)
               : "memory");
}
__device__ __forceinline__ void async_gather_b128(uint32_t lds, const void* g) {
  asm volatile("global_load_async_to_lds_b128 %0, %1, off"
               :: "v"(lds), "v"((unsigned long long)(uintptr_t)g)
               : "memory");
}
__device__ __forceinline__ void wait_async_all() {
  asm volatile("s_wait_asynccnt 0" ::: "memory");
}

__global__ __launch_bounds__(TPB) void qnms_fused(
    const float* __restrict__ r, const float* __restrict__ ac,
    const float* __restrict__ bc, const float* __restrict__ av,
    const float* __restrict__ bv, const int* __restrict__ H,
    float* __restrict__ out) {
  __shared__ unsigned short s_cols[EDGES];       // column index of each edge
  __shared__ float s_E[EDGES];                   // messages; later aliased as H bitmask
  __shared__ float s_sumE[N_COLS];               // per-column accumulators
  __shared__ __align__(16) float s_r[N_COLS];    // channel values for this batch
  __shared__ float s_min1[M_ROWS];
  __shared__ float s_min2[M_ROWS];
  __shared__ float s_sp[M_ROWS];                 // row sign product
  __shared__ int   s_idx[M_ROWS];                // column of min1
  __shared__ float s_tab[4 * EDGES];             // alpha/beta tables; later tail tile

  float* s_tAc = s_tab + 0 * EDGES;
  float* s_tBc = s_tab + 1 * EDGES;
  float* s_tAv = s_tab + 2 * EDGES;
  float* s_tBv = s_tab + 3 * EDGES;

  const int tid = threadIdx.x;
  const int b = blockIdx.x;

  // Stage this batch's r row: 16B per lane through the async DMA engine.
  for (int n = tid * 4; n < N_COLS; n += TPB * 4)
    async_gather_b128(lds_off(&s_r[n]), &r[(size_t)b * N_COLS + n]);

  // Build per-row edge column lists (exactly DEG ones per row by construction).
  if (tid < M_ROWS) {
    const int* hrow = H + (size_t)tid * N_COLS;
    int cnt = 0;
    for (int n = 0; n < N_COLS; ++n)
      if (hrow[n] != 0 && cnt < DEG) s_cols[tid * DEG + (cnt++)] = (unsigned short)n;
  }
  wait_async_all();
  __syncthreads();

  // Kick off iteration-0 table gathers (edge-indexed async gathers to LDS).
  for (int e = tid; e < EDGES; e += TPB) {
    int m = e / DEG;
    int n = s_cols[e];
    size_t gi = ((size_t)m * N_COLS + n) * N_ITERS;  // + it (it = 0)
    async_gather_b32(lds_off(&s_tAc[e]), &ac[gi]);
    async_gather_b32(lds_off(&s_tBc[e]), &bc[gi]);
    async_gather_b32(lds_off(&s_tAv[e]), &av[gi]);
    async_gather_b32(lds_off(&s_tBv[e]), &bv[gi]);
  }

  // E = H * r at edge positions.
  for (int e = tid; e < EDGES; e += TPB)
    s_E[e] = s_r[s_cols[e]];
  __syncthreads();

  for (int it = 0; it < N_ITERS; ++it) {
    // Per-row top-2 min of |E| + sign product; zero the column sums.
    for (int n = tid; n < N_COLS; n += TPB) s_sumE[n] = 0.0f;
    if (tid < M_ROWS) {
      float m1 = __builtin_inff(), m2 = __builtin_inff(), sp = 1.0f;
      int c1 = -1;
      for (int e = 0; e < DEG; ++e) {
        float v = s_E[tid * DEG + e];
        float a = fabsf(v);
        int n = s_cols[tid * DEG + e];
        if (a < m1) { m2 = m1; m1 = a; c1 = n; }
        else if (a < m2) { m2 = a; }
        sp *= sgnf(v);
      }
      s_min1[tid] = m1; s_min2[tid] = m2; s_sp[tid] = sp; s_idx[tid] = c1;
    }
    wait_async_all();   // this iteration's alpha/beta tables now resident
    __syncthreads();

    // Check-node update + column scatter-add (ds_add_f32).
    for (int e = tid; e < EDGES; e += TPB) {
      int m = e / DEG;
      int n = s_cols[e];
      float Eold = s_E[e];
      float Eabs = (n == s_idx[m]) ? s_min2[m] : s_min1[m];
      float val = s_sp[m] * sgnf(Eold) * fmaxf(s_tAc[e] * Eabs + s_tBc[e], 0.0f);
      float Enew = qsoft(val);
      s_E[e] = Enew;
      atomicAdd(&s_sumE[n], Enew);
    }
    __syncthreads();

    // Variable-node update (edge positions; Hf == 1 there).
    for (int e = tid; e < EDGES; e += TPB) {
      int n = s_cols[e];
      float rv = s_r[n];
      float Ev = rv + s_sumE[n] - s_E[e];
      s_E[e] = qsoft(fmaxf(s_tAv[e] * Ev + s_tBv[e], 0.0f) + rv);
    }
    __syncthreads();

    // Prefetch next iteration's tables; overlaps next row-stats phase.
    if (it + 1 < N_ITERS) {
      for (int e = tid; e < EDGES; e += TPB) {
        int m = e / DEG;
        int n = s_cols[e];
        size_t gi = ((size_t)m * N_COLS + n) * N_ITERS + (it + 1);
        async_gather_b32(lds_off(&s_tAc[e]), &ac[gi]);
        async_gather_b32(lds_off(&s_tBc[e]), &bc[gi]);
        async_gather_b32(lds_off(&s_tAv[e]), &av[gi]);
        async_gather_b32(lds_off(&s_tBv[e]), &bv[gi]);
      }
    }
  }

  // Edge contribution into s_sumE.
  for (int n = tid; n < N_COLS; n += TPB) s_sumE[n] = 0.0f;
  __syncthreads();
  for (int e = tid; e < EDGES; e += TPB)
    atomicAdd(&s_sumE[s_cols[e]], s_E[e]);
  __syncthreads();

  // s_E is dead now: alias it as a per-column H bitmask (576 x 5 words).
  uint32_t* s_hb = reinterpret_cast<uint32_t*>(s_E);
  for (int w = tid; w < N_COLS * 5; w += TPB) s_hb[w] = 0u;
  __syncthreads();
  for (int e = tid; e < EDGES; e += TPB) {
    int m = e / DEG;
    int n = s_cols[e];
    atomicOr(&s_hb[n * 5 + (m >> 5)], 1u << (m & 31));
  }
  __syncthreads();

  // Dense non-edge tail: _Q(relu(beta_v[m,n,last]) + r), summed over m.
  // Stage beta_v[:, chunk, 2] column-major into the (now dead) table buffer
  // via async DMA, then consume from LDS. 6 chunks of 96 columns.
  for (int c0 = 0; c0 < N_COLS; c0 += CHUNK) {
    for (int idx = tid; idx < CHUNK * M_ROWS; idx += TPB) {
      int cl = idx / M_ROWS;          // column within chunk
      int m  = idx - cl * M_ROWS;     // row
      size_t gi = ((size_t)m * N_COLS + (c0 + cl)) * N_ITERS + (N_ITERS - 1);
      async_gather_b32(lds_off(&s_tab[idx]), &bv[gi]);
    }
    wait_async_all();
    __syncthreads();

    for (int w = tid; w < CHUNK * NSEG; w += TPB) {
      int cl  = w >> 3;               // column within chunk
      int seg = w & (NSEG - 1);       // 18-row segment
      int n = c0 + cl;
      float rv = s_r[n];
      float acc = 0.0f;
      int m0 = seg * (M_ROWS / NSEG);
      for (int m = m0; m < m0 + (M_ROWS / NSEG); ++m) {
        bool edge = (s_hb[n * 5 + (m >> 5)] >> (m & 31)) & 1u;
        if (!edge)
          acc += qsoft(fmaxf(s_tab[cl * M_ROWS + m], 0.0f) + rv);
      }
      atomicAdd(&s_sumE[n], acc);
    }
    __syncthreads();   // chunk tile fully consumed before next overwrite
  }

  // out = r + (edge sum + dense tail), both accumulated in s_sumE.
  for (int n = tid; n < N_COLS; n += TPB)
    out[(size_t)b * N_COLS + n] = s_r[n] + s_sumE[n];
}

extern "C" void kernel_launch(void* const* d_in, const int* in_sizes, int n_in,
                              void* d_out, int out_size, void* d_ws, size_t ws_size,
                              hipStream_t stream) {
  (void)in_sizes; (void)n_in; (void)out_size; (void)d_ws; (void)ws_size;
  const float* r  = (const float*)d_in[0];
  const float* ac = (const float*)d_in[1];
  const float* bc = (const float*)d_in[2];
  const float* av = (const float*)d_in[3];
  const float* bv = (const float*)d_in[4];
  const int*   H  = (const int*)d_in[5];
  float* out = (float*)d_out;
  qnms_fused<<<BATCH, TPB, 0, stream>>>(r, ac, bc, av, bv, H, out);
}